// GCN_30966714204810
// MI455X (gfx1250) — compile-verified
//
#include <hip/hip_runtime.h>

#define N_NODES 100000
#define N_EDGES 600000
#define HD 128

typedef __attribute__((ext_vector_type(2))) float v2f;
typedef __attribute__((ext_vector_type(8))) float v8f;

// ---------------------------------------------------------------------------
// h[i, :] = embed[node_id[i], :]   (one float4 per thread, 32 lanes per row)
// ---------------------------------------------------------------------------
__global__ __launch_bounds__(256) void gather_embed(
    const int* __restrict__ node_id, const float* __restrict__ embed,
    float* __restrict__ h) {
  int t   = blockIdx.x * blockDim.x + threadIdx.x;
  int row = t >> 5;
  int c4  = (t & 31) << 2;
  if (row < N_NODES) {
    int srow = node_id[row];
    float4 v = *(const float4*)(embed + (size_t)srow * HD + c4);
    *(float4*)(h + (size_t)row * HD + c4) = v;
  }
}

// ---------------------------------------------------------------------------
// m = (h @ W) * norm  using V_WMMA_F32_16X16X4_F32
// Block = 256 threads = 8 waves; each wave: 16 rows x 128 cols.
//
// W is staged into LDS in K-pair-interleaved form:
//   Wl2[p*256 + 2n + 0] = W[2p  ][n]
//   Wl2[p*256 + 2n + 1] = W[2p+1][n]
// so a lane's B fragment {W[k][n], W[k+1][n]} is ONE aligned ds_load_b64
// directly into the even-aligned WMMA operand pair (no register repacking).
//
// fp32 fragment layout (ISA 7.12.2):
//   A 16x4:  lanes 0-15 -> M=lane, VGPR0=K0,VGPR1=K1; lanes 16-31 -> K2,K3
//   B 4x16:  lanes 0-15 -> N=lane, VGPR0=K0,VGPR1=K1; lanes 16-31 -> K2,K3
//   C/D 16x16: VGPR v, lane L -> row v+(L>=16?8:0), col L&15
// ---------------------------------------------------------------------------
__global__ __launch_bounds__(256) void gemm_norm_wmma(
    const float* __restrict__ h, const float* __restrict__ W,
    const float* __restrict__ norm, float* __restrict__ m) {
  __shared__ float Wl2[HD * HD];   // 64 KB, pair-interleaved layout

  int tid = threadIdx.x;
  // Stage + swizzle: i indexes (pair p, n-pair). 4096 float4 stores total.
  for (int i = tid; i < (HD * HD) / 4; i += 256) {
    int p = i >> 6;               // K-pair index 0..63
    int n = (i & 63) << 1;        // n, n+1
    int k = p << 1;
    float2 r0 = *(const float2*)(W + (size_t)k * HD + n);        // W[k][n..n+1]
    float2 r1 = *(const float2*)(W + (size_t)(k + 1) * HD + n);  // W[k+1][n..n+1]
    float4 o  = make_float4(r0.x, r1.x, r0.y, r1.y);
    *(float4*)(Wl2 + p * 256 + n * 2) = o;
  }
  __syncthreads();

  int wave    = tid >> 5;
  int lane    = tid & 31;
  int lane16  = lane & 15;
  int pOff    = lane >> 4;          // 0 or 1 (K half within the 4-wide step)
  int rowBase = (blockIdx.x * 8 + wave) * 16;
  int rowA    = rowBase + lane16;
  int rowAc   = rowA < N_NODES ? rowA : (N_NODES - 1);

  v8f acc[8] = {};
  const float* hrow = h + (size_t)rowAc * HD;

  for (int k0 = 0; k0 < HD; k0 += 4) {
    v2f a = *(const v2f*)(hrow + k0 + (pOff << 1));       // K=k..k+1, this half
    const float* bbase = Wl2 + ((k0 >> 1) + pOff) * 256;
#pragma unroll
    for (int t = 0; t < 8; ++t) {
      int n = t * 16 + lane16;
      v2f b = *(const v2f*)(bbase + n * 2);               // one ds_load_b64
      acc[t] = __builtin_amdgcn_wmma_f32_16x16x4_f32(
          false, a, false, b, (short)0, acc[t], false, false);
    }
  }

  int rhalf = pOff << 3;            // 0 or 8
#pragma unroll
  for (int v = 0; v < 8; ++v) {
    int r = rowBase + v + rhalf;
    if (r < N_NODES) {
      float nr = norm[r];
#pragma unroll
      for (int t = 0; t < 8; ++t)
        m[(size_t)r * HD + t * 16 + lane16] = acc[t][v] * nr;
    }
  }
}

// ---------------------------------------------------------------------------
__global__ __launch_bounds__(256) void zero_f4(float4* __restrict__ p, int n4) {
  int t = blockIdx.x * blockDim.x + threadIdx.x;
  if (t < n4) p[t] = make_float4(0.f, 0.f, 0.f, 0.f);
}

// ---------------------------------------------------------------------------
// agg[dst[e], :] += m[src[e], :]   — one wave per edge, float4 per lane,
// hardware fp32 atomics (no-return form since result is discarded).
// ---------------------------------------------------------------------------
__global__ __launch_bounds__(256) void scatter_edges(
    const float* __restrict__ m, const int* __restrict__ src,
    const int* __restrict__ dst, float* __restrict__ agg) {
  int wave = threadIdx.x >> 5;
  int lane = threadIdx.x & 31;
  int e = blockIdx.x * 8 + wave;
  if (e >= N_EDGES) return;
  int s = src[e];
  int d = dst[e];
  float4 v    = *(const float4*)(m + (size_t)s * HD + lane * 4);
  float* dstp = agg + (size_t)d * HD + lane * 4;
  (void)__hip_atomic_fetch_add(dstp + 0, v.x, __ATOMIC_RELAXED, __HIP_MEMORY_SCOPE_AGENT);
  (void)__hip_atomic_fetch_add(dstp + 1, v.y, __ATOMIC_RELAXED, __HIP_MEMORY_SCOPE_AGENT);
  (void)__hip_atomic_fetch_add(dstp + 2, v.z, __ATOMIC_RELAXED, __HIP_MEMORY_SCOPE_AGENT);
  (void)__hip_atomic_fetch_add(dstp + 3, v.w, __ATOMIC_RELAXED, __HIP_MEMORY_SCOPE_AGENT);
}

// ---------------------------------------------------------------------------
// out = LeakyReLU_0.2(agg * norm + b)   (in-place safe: same-thread RMW)
// ---------------------------------------------------------------------------
__global__ __launch_bounds__(256) void epilogue_leaky(
    const float* __restrict__ agg, const float* __restrict__ norm,
    const float* __restrict__ bias, float* __restrict__ out) {
  int t   = blockIdx.x * blockDim.x + threadIdx.x;
  int row = t >> 5;
  int c4  = (t & 31) << 2;
  if (row < N_NODES) {
    float nr = norm[row];
    float4 a = *(const float4*)(agg + (size_t)row * HD + c4);
    float4 b = *(const float4*)(bias + c4);
    float4 n;
    n.x = a.x * nr + b.x;
    n.y = a.y * nr + b.y;
    n.z = a.z * nr + b.z;
    n.w = a.w * nr + b.w;
    n.x = n.x > 0.f ? n.x : 0.2f * n.x;
    n.y = n.y > 0.f ? n.y : 0.2f * n.y;
    n.z = n.z > 0.f ? n.z : 0.2f * n.z;
    n.w = n.w > 0.f ? n.w : 0.2f * n.w;
    *(float4*)(out + (size_t)row * HD + c4) = n;
  }
}

// ---------------------------------------------------------------------------
extern "C" void kernel_launch(void* const* d_in, const int* in_sizes, int n_in,
                              void* d_out, int out_size, void* d_ws, size_t ws_size,
                              hipStream_t stream) {
  const int*   node_id = (const int*)d_in[0];
  const int*   src     = (const int*)d_in[1];
  const int*   dst     = (const int*)d_in[2];
  const float* norm    = (const float*)d_in[3];
  const float* embed   = (const float*)d_in[4];
  const float* W1      = (const float*)d_in[5];
  const float* b1      = (const float*)d_in[6];
  const float* W2      = (const float*)d_in[7];
  const float* b2      = (const float*)d_in[8];
  float* out = (float*)d_out;

  const size_t feat = (size_t)N_NODES * HD;       // 12.8M floats
  float* bufH = (float*)d_ws;                     // node features
  float* bufM = bufH + feat;                      // post-GEMM messages
  float* agg  = out;                              // d_out doubles as agg buffer

  dim3 blk(256);
  const int gElem    = (int)((feat / 4 + 255) / 256);   // 12500 (float4 grids)
  const int gGemm    = (N_NODES + 127) / 128;           // 782
  const int gScatter = (N_EDGES + 7) / 8;               // 75000

  gather_embed<<<gElem, blk, 0, stream>>>(node_id, embed, bufH);

  // ---- layer 1 ----
  gemm_norm_wmma<<<gGemm, blk, 0, stream>>>(bufH, W1, norm, bufM);
  zero_f4<<<gElem, blk, 0, stream>>>((float4*)agg, (int)(feat / 4));
  scatter_edges<<<gScatter, blk, 0, stream>>>(bufM, src, dst, agg);
  epilogue_leaky<<<gElem, blk, 0, stream>>>(agg, norm, b1, bufH);

  // ---- layer 2 ----
  gemm_norm_wmma<<<gGemm, blk, 0, stream>>>(bufH, W2, norm, bufM);
  zero_f4<<<gElem, blk, 0, stream>>>((float4*)agg, (int)(feat / 4));
  scatter_edges<<<gScatter, blk, 0, stream>>>(bufM, src, dst, agg);
  epilogue_leaky<<<gElem, blk, 0, stream>>>(agg, norm, b2, out);  // in-place
}